// SparseQuanvLayer_66142496358522
// MI455X (gfx1250) — compile-verified
//
#include <hip/hip_runtime.h>
#include <stdint.h>

typedef float v2f __attribute__((ext_vector_type(2)));
typedef float v8f __attribute__((ext_vector_type(8)));

// Packed op encoding: bit0: 1=rotation, 0=cnot
//  rotation: kind=(e>>2)&3, wire=(e>>4)&3, param_idx=(e>>8)&255
//  cnot:     ctrl=(e>>2)&3, tgt=(e>>4)&3
struct OpList { int n; int op[32]; };

// ---------------------------------------------------------------------------
// Setup kernel: build M_w = Re(U^H Z_w U) for w=0..3  (4 x 16x16 f32 -> d_ws)
// Wire w corresponds to state bit (3-w): s = a*8+b*4+c*2+d, a=wire0.
// ---------------------------------------------------------------------------
__global__ void build_measure_mats(const float* __restrict__ params,
                                   float* __restrict__ M, OpList ops) {
  if (threadIdx.x != 0 || blockIdx.x != 0) return;
  float Ur[16][16], Ui[16][16];
  for (int r = 0; r < 16; ++r)
    for (int c = 0; c < 16; ++c) { Ur[r][c] = (r == c) ? 1.f : 0.f; Ui[r][c] = 0.f; }

  for (int o = 0; o < ops.n; ++o) {
    int e = ops.op[o];
    if (e & 1) {  // single-qubit rotation: U = G(w) * U
      int kind = (e >> 2) & 3, w = (e >> 4) & 3, pidx = (e >> 8) & 255;
      float th = params[pidx] * 0.5f;
      float cc = cosf(th), ss = sinf(th);
      // 2x2 complex gate
      float g00r, g00i, g01r, g01i, g10r, g10i, g11r, g11i;
      if (kind == 0) {        // RX
        g00r = cc; g00i = 0;  g01r = 0;  g01i = -ss;
        g10r = 0;  g10i = -ss; g11r = cc; g11i = 0;
      } else if (kind == 1) { // RY
        g00r = cc; g00i = 0;  g01r = -ss; g01i = 0;
        g10r = ss; g10i = 0;  g11r = cc;  g11i = 0;
      } else {                // RZ
        g00r = cc; g00i = -ss; g01r = 0; g01i = 0;
        g10r = 0;  g10i = 0;   g11r = cc; g11i = ss;
      }
      int mask = 1 << (3 - w);
      for (int s0 = 0; s0 < 16; ++s0) {
        if (s0 & mask) continue;
        int s1 = s0 | mask;
        for (int c = 0; c < 16; ++c) {
          float ar = Ur[s0][c], ai = Ui[s0][c];
          float br = Ur[s1][c], bi = Ui[s1][c];
          Ur[s0][c] = g00r*ar - g00i*ai + g01r*br - g01i*bi;
          Ui[s0][c] = g00r*ai + g00i*ar + g01r*bi + g01i*br;
          Ur[s1][c] = g10r*ar - g10i*ai + g11r*br - g11i*bi;
          Ui[s1][c] = g10r*ai + g10i*ar + g11r*bi + g11i*br;
        }
      }
    } else {  // CNOT(ctrl, tgt): rows with ctrl-bit=1 swap on tgt-bit
      int cw = (e >> 2) & 3, tw = (e >> 4) & 3;
      int mc = 1 << (3 - cw), mt = 1 << (3 - tw);
      for (int s = 0; s < 16; ++s) {
        if ((s & mc) && !(s & mt)) {
          int s2 = s | mt;
          for (int c = 0; c < 16; ++c) {
            float t;
            t = Ur[s][c]; Ur[s][c] = Ur[s2][c]; Ur[s2][c] = t;
            t = Ui[s][c]; Ui[s][c] = Ui[s2][c]; Ui[s2][c] = t;
          }
        }
      }
    }
  }

  // M_w[m][n] = sum_s sign_w(s) * Re(conj(U[s][m]) U[s][n])
  for (int w = 0; w < 4; ++w) {
    int mask = 1 << (3 - w);
    for (int m = 0; m < 16; ++m)
      for (int n = 0; n < 16; ++n) {
        float acc = 0.f;
        for (int s = 0; s < 16; ++s) {
          float sg = (s & mask) ? -1.f : 1.f;
          acc += sg * (Ur[s][m] * Ur[s][n] + Ui[s][m] * Ui[s][n]);
        }
        M[w * 256 + m * 16 + n] = acc;
      }
  }
}

// ---------------------------------------------------------------------------
// Main kernel: 16 sites per wave. Psi (16x16) as B, M_w as A,
// Y = M_w x Psi via 4x v_wmma_f32_16x16x4_f32, z_w = colwise <Psi, Y>.
// ---------------------------------------------------------------------------
__global__ __launch_bounds__(256) void quanv_wmma_kernel(
    const float* __restrict__ x, const float* __restrict__ M,
    float* __restrict__ out) {
  const int lane = threadIdx.x & 31;
  const int wave = blockIdx.x * (blockDim.x >> 5) + (threadIdx.x >> 5);
  const int r = lane & 15;   // column (site) index within tile
  const int h = lane >> 4;   // half-wave
  const int row = wave * 16 + r;

  // row -> (b, i, j, k) of shape (16,32,32,32)
  const int kk = row & 31;
  const int jj = (row >> 5) & 31;
  const int ii = (row >> 10) & 31;
  const int bb = row >> 15;

  const float* xb = x + ((((size_t)bb * 64 + 2 * ii) * 64 + 2 * jj) * 64 + 2 * kk);
  const float p0 = xb[0];              // [2i,   2j,   2k  ]
  const float p1 = xb[64 * 64 + 64];   // [2i+1, 2j+1, 2k  ]
  const float p2 = xb[64 + 1];         // [2i,   2j+1, 2k+1]
  const float p3 = xb[64 * 64 + 1];    // [2i+1, 2j,   2k+1]

  // angle = (pi/2)*p -> use sincospif(p/2)
  float sn0, cs0, sn1, cs1, sn2, cs2, sn3, cs3;
  sincospif(0.5f * p0, &sn0, &cs0);
  sincospif(0.5f * p1, &sn1, &cs1);
  sincospif(0.5f * p2, &sn2, &cs2);
  sincospif(0.5f * p3, &sn3, &cs3);
  const float a0[2] = {cs0, sn0}, a1[2] = {cs1, sn1};
  const float a2[2] = {cs2, sn2}, a3[2] = {cs3, sn3};

  float psi[16];
#pragma unroll
  for (int s = 0; s < 16; ++s)
    psi[s] = a0[(s >> 3) & 1] * a1[(s >> 2) & 1] * a2[(s >> 1) & 1] * a3[s & 1];

  // B fragments (4x16, K-chunk q): VGPR0 = K(4q+2h), VGPR1 = K(4q+2h+1)
  v2f bfrag[4];
#pragma unroll
  for (int q = 0; q < 4; ++q) {
    bfrag[q].x = psi[4 * q + 2 * h];
    bfrag[q].y = psi[4 * q + 2 * h + 1];
  }

  float z[4];
#pragma unroll
  for (int w = 0; w < 4; ++w) {
    const float* Mw = M + w * 256 + r * 16;  // A-matrix row m = lane&15
    v8f acc = {};
#pragma unroll
    for (int q = 0; q < 4; ++q) {
      v2f afrag;
      afrag.x = Mw[4 * q + 2 * h];
      afrag.y = Mw[4 * q + 2 * h + 1];
      acc = __builtin_amdgcn_wmma_f32_16x16x4_f32(
          false, afrag, false, bfrag[q], (short)0, acc, false, false);
    }
    // z_w[r] = sum_s psi[s]*Y[s][r]; lane holds Y rows v+8h in acc[v]
    float partial = 0.f;
#pragma unroll
    for (int v = 0; v < 8; ++v) partial += psi[v + 8 * h] * acc[v];
    z[w] = partial + __shfl_xor(partial, 16, 32);
  }

  if (h == 0) {
    float4 o; o.x = z[0]; o.y = z[1]; o.z = z[2]; o.w = z[3];
    reinterpret_cast<float4*>(out)[row] = o;
  }
}

// ---------------------------------------------------------------------------
// Host: replicate numpy default_rng(42) stream to rebuild the static op list.
// ---------------------------------------------------------------------------
typedef unsigned __int128 u128;

namespace nprng {
struct Pcg64 {
  u128 state, inc;
  int has32; uint32_t cached;
  void step() {
    const u128 mult = ((u128)2549297995355413924ULL << 64) | 4865540595714422341ULL;
    state = state * mult + inc;
  }
  uint64_t next64() {
    step();
    uint64_t hi = (uint64_t)(state >> 64), lo = (uint64_t)state;
    uint64_t v = hi ^ lo;
    unsigned rot = (unsigned)(state >> 122);
    return (v >> rot) | (v << ((64u - rot) & 63u));
  }
  uint32_t next32() {
    if (has32) { has32 = 0; return cached; }
    uint64_t n = next64();
    has32 = 1; cached = (uint32_t)(n >> 32);
    return (uint32_t)n;
  }
  double rnd() { return (double)(next64() >> 11) * (1.0 / 9007199254740992.0); }
  uint64_t lemire(uint64_t rng) {  // inclusive max
    uint64_t excl = rng + 1;
    u128 m = (u128)next64() * excl;
    uint64_t left = (uint64_t)m;
    if (left < excl) {
      uint64_t thr = (uint64_t)(0ULL - excl) % excl;
      while (left < thr) { m = (u128)next64() * excl; left = (uint64_t)m; }
    }
    return (uint64_t)(m >> 64);
  }
  uint64_t interval(uint64_t mx) {  // masked rejection, 32-bit path (mx small)
    if (mx == 0) return 0;
    uint64_t mask = mx;
    mask |= mask >> 1; mask |= mask >> 2; mask |= mask >> 4;
    mask |= mask >> 8; mask |= mask >> 16; mask |= mask >> 32;
    uint64_t v;
    while ((v = (uint64_t)(next32() & (uint32_t)mask)) > mx) {}
    return v;
  }
};

static void seedseq42(uint64_t out64[4]) {
  const uint32_t XS = 16;
  uint32_t pool[4];
  uint32_t hc = 0x43b0d7e5u;
  auto hashmix = [&](uint32_t v) {
    v ^= hc; hc *= 0x931e8875u; v *= hc; v ^= v >> XS; return v;
  };
  auto mix = [&](uint32_t xx, uint32_t yy) {
    uint32_t rr = xx * 0xca01f9ddu - yy * 0x4973f715u;
    rr ^= rr >> XS; return rr;
  };
  for (int i = 0; i < 4; ++i) pool[i] = hashmix(i == 0 ? 42u : 0u);
  for (int s = 0; s < 4; ++s)
    for (int d = 0; d < 4; ++d)
      if (s != d) pool[d] = mix(pool[d], hashmix(pool[s]));
  uint32_t w32[8];
  uint32_t hc2 = 0x8b51f9ddu;
  for (int i = 0; i < 8; ++i) {
    uint32_t v = pool[i % 4];
    v ^= hc2; hc2 *= 0x58f38dedu; v *= hc2; v ^= v >> XS;
    w32[i] = v;
  }
  for (int i = 0; i < 4; ++i)
    out64[i] = (uint64_t)w32[2 * i] | ((uint64_t)w32[2 * i + 1] << 32);
}

static Pcg64 make_rng42() {
  uint64_t s[4];
  seedseq42(s);
  u128 initstate = ((u128)s[0] << 64) | s[1];
  u128 initseq = ((u128)s[2] << 64) | s[3];
  Pcg64 r;
  r.has32 = 0; r.cached = 0;
  r.state = 0;
  r.inc = (initseq << 1) | 1;
  r.step();
  r.state += initstate;
  r.step();
  return r;
}
}  // namespace nprng

static OpList build_ops() {
  nprng::Pcg64 rng = nprng::make_rng42();
  OpList ops; ops.n = 0;
  int i = 0;
  while (i < 4 && ops.n < 32) {
    if (rng.rnd() > 0.3) {
      int kind = (int)rng.lemire(2);   // integers(0,3)
      int w = (int)rng.lemire(3);      // integers(0,4)
      ops.op[ops.n++] = 1 | (kind << 2) | (w << 4) | (i << 8);
      ++i;
    } else {
      int arr[4] = {0, 1, 2, 3};       // permutation(4): Fisher-Yates
      for (int a = 3; a >= 1; --a) {
        int j = (int)rng.interval((uint64_t)a);
        int t = arr[a]; arr[a] = arr[j]; arr[j] = t;
      }
      ops.op[ops.n++] = 0 | (arr[0] << 2) | (arr[1] << 4);
    }
  }
  return ops;
}

extern "C" void kernel_launch(void* const* d_in, const int* in_sizes, int n_in,
                              void* d_out, int out_size, void* d_ws, size_t ws_size,
                              hipStream_t stream) {
  const float* x = (const float*)d_in[0];        // (16,64,64,64) f32
  const float* params = (const float*)d_in[1];   // (1,4) f32
  float* out = (float*)d_out;                    // (16,32,32,32,4) f32
  float* M = (float*)d_ws;                       // 4*16*16 f32 = 4 KB

  OpList ops = build_ops();
  build_measure_mats<<<1, 1, 0, stream>>>(params, M, ops);

  const int rows = 16 * 32 * 32 * 32;  // 524288 sites
  const int waves = rows / 16;         // 16 sites per wave
  const int blocks = waves / 8;        // 256 threads = 8 waves per block
  quanv_wmma_kernel<<<blocks, 256, 0, stream>>>(x, M, out);
}